// HandAwareModelDecoder_22814866276635
// MI455X (gfx1250) — compile-verified
//
#include <hip/hip_runtime.h>
#include <hip/hip_bf16.h>

typedef __attribute__((ext_vector_type(2))) float v2f;
typedef __attribute__((ext_vector_type(8))) float v8f;
typedef __attribute__((ext_vector_type(4))) unsigned int u32x4;
typedef __attribute__((ext_vector_type(8))) int i32x8;
typedef __attribute__((ext_vector_type(4))) int i32x4;

#define NTOK   4096      // 32 * 128
#define DIM    5376
#define NOUT   70        // 45+3+10+9+2+1
#define PSTRIDE 80       // padded params stride in workspace
#define NVERT  778
#define KC     128       // K-chunk staged in LDS per TDM transfer
#define NCHUNK (DIM / KC)

// ---------------------------------------------------------------------------
// Kernel 1: precompute Jbase = J_reg @ template  (16x3)
//           and        JS    = J_reg @ shapedirs (16x3x10)
// ---------------------------------------------------------------------------
__global__ void precompute_jreg(const float* __restrict__ J_reg,      // (16,778)
                                const float* __restrict__ templ,      // (778,3)
                                const float* __restrict__ shapedirs,  // (778,3,10)
                                float* __restrict__ jbase,            // (16,3)
                                float* __restrict__ js)               // (16,3,10)
{
    int tid = threadIdx.x;
    if (tid >= 16 * 3 * 11) return;
    int j = tid / 33;
    int r = tid % 33;
    int c = r / 11;
    int k = r % 11;
    const float* jr = J_reg + j * NVERT;
    float acc = 0.0f;
    if (k == 0) {
        for (int v = 0; v < NVERT; ++v)
            acc += jr[v] * templ[v * 3 + c];
        jbase[j * 3 + c] = acc;
    } else {
        int kk = k - 1;
        for (int v = 0; v < NVERT; ++v)
            acc += jr[v] * shapedirs[v * 30 + c * 10 + kk];
        js[(j * 3 + c) * 10 + kk] = acc;
    }
}

// ---------------------------------------------------------------------------
// TDM: stage a 70 x KC fp32 slab of W (row-major, row stride 5376) into LDS.
// D# per CDNA5 ISA ch.8: group0 = {count/type/lds_addr/global_addr},
// group1 = {data_size, tensor dims, tile dims, strides}. 2D tile, no pad,
// no multicast (workgroup_mask = 0), no iterate. Groups 2/3 unused (2D).
// This toolchain exposes the 6-arg builtin (extra trailing 8-dword group).
// ---------------------------------------------------------------------------
__device__ __forceinline__ void tdm_load_w(const float* Wg, unsigned kc,
                                           unsigned lds_byte_off) {
    unsigned long long ga =
        (unsigned long long)(uintptr_t)Wg + (unsigned long long)kc * 4ull;
    u32x4 g0;
    g0[0] = 1u;                                   // count=1 (valid user D#)
    g0[1] = lds_byte_off;                         // lds_addr (bytes)
    g0[2] = (unsigned)(ga & 0xFFFFFFFFull);       // global_addr[31:0]
    g0[3] = (unsigned)((ga >> 32) & 0x01FFFFFFull) | 0x80000000u; // [56:32]|type=2
    i32x8 g1;
    g1[0] = 0x00020000;                           // data_size=2 -> 4 bytes
    g1[1] = (int)((unsigned)DIM << 16);           // tensor_dim0[15:0] @ bits63:48
    g1[2] = (int)((unsigned)NOUT << 16);          // dim0 hi=0; tensor_dim1=70
    g1[3] = (int)((unsigned)KC << 16);            // dim1 hi=0; tile_dim0=KC
    g1[4] = NOUT;                                 // tile_dim1=70, tile_dim2=0
    g1[5] = DIM;                                  // tensor_dim0_stride = 5376
    g1[6] = 0;
    g1[7] = 0;
    i32x4 z4 = {0, 0, 0, 0};
    i32x8 z8 = {0, 0, 0, 0, 0, 0, 0, 0};
    __builtin_amdgcn_tensor_load_to_lds(g0, g1, z4, z4, z8, 0);
}

// ---------------------------------------------------------------------------
// Kernel 2: params = x @ W^T + b via V_WMMA_F32_16X16X4_F32 (wave32).
// Block = 5 waves; wave w owns N-tile w (cols 16w..16w+15) of one 16-row
// M-tile. W slabs are TDM-staged into double-buffered LDS and shared by all
// 5 waves; x fragments stream from global (HBM read exactly once; L2 absorbs
// the 5x intra-block re-reads).
//
// A (16x4 f32): lanes 0-15 -> M=lane, VGPRs K+0,K+1; lanes16-31 -> K+2,K+3.
// B (4x16 f32): lanes 0-15 -> N=lane, VGPRs K+0,K+1; lanes16-31 -> K+2,K+3.
// C/D (16x16 f32): VGPR r, lanes 0-15 -> M=r; lanes 16-31 -> M=r+8.
// ---------------------------------------------------------------------------
__global__ __launch_bounds__(160)
void gemm_wmma(const float* __restrict__ x,     // (4096, 5376)
               const float* __restrict__ W,     // (70, 5376)
               const float* __restrict__ bias,  // (70,)
               float* __restrict__ out)         // (4096, 80) padded
{
    __shared__ float lw[2][NOUT * KC];           // 2 x 35840 B

    const int wv   = threadIdx.x >> 5;           // wave id = N-tile (0..4)
    const int lane = threadIdx.x & 31;
    const int m0   = blockIdx.x * 16;

    const int l15  = lane & 15;
    const int kOff = (lane >> 4) * 2;            // 0 or 2

    const int rowA = m0 + l15;
    int colL = wv * 16 + l15;                    // W row / output col
    if (colL > NOUT - 1) colL = NOUT - 1;        // clamp; junk cols not stored

    const float* xrow = x + (size_t)rowA * DIM + kOff;

    const unsigned lds0 = (unsigned)(uintptr_t)(&lw[0][0]);
    const unsigned lds1 = (unsigned)(uintptr_t)(&lw[1][0]);

    if (wv == 0) tdm_load_w(W, 0, lds0);

    v8f acc = {};
    for (int c = 0; c < NCHUNK; ++c) {
        if (wv == 0) {
            if (c + 1 < NCHUNK) {
                tdm_load_w(W, (unsigned)((c + 1) * KC),
                           ((c + 1) & 1) ? lds1 : lds0);
                __builtin_amdgcn_s_wait_tensorcnt(1);  // oldest (chunk c) done
            } else {
                __builtin_amdgcn_s_wait_tensorcnt(0);
            }
        }
        __syncthreads();                          // LDS slab c visible to all

        const float* bptr = &lw[c & 1][0] + (size_t)colL * KC + kOff;
        const float* ax   = xrow + (size_t)c * KC;
#pragma unroll 8
        for (int kk = 0; kk < KC; kk += 4) {
            __builtin_prefetch(ax + kk + KC, 0, 1);   // next slab of x
            v2f a  = *(const v2f*)(ax + kk);
            v2f bb = *(const v2f*)(bptr + kk);
            acc = __builtin_amdgcn_wmma_f32_16x16x4_f32(
                false, a, false, bb, (short)0, acc, false, false);
        }
        __syncthreads();                          // done reading before rewrite
    }

    const int col     = wv * 16 + l15;
    const int rowBase = m0 + ((lane >> 4) << 3);  // +8 for upper half-wave
    if (col < NOUT) {
        const float bv = bias[col];
#pragma unroll
        for (int r = 0; r < 8; ++r)
            out[(size_t)(rowBase + r) * PSTRIDE + col] = acc[r] + bv;
    }
}

// ---------------------------------------------------------------------------
// Kernel 3: per-token decode. One 128-thread block per token.
// ---------------------------------------------------------------------------
__device__ __forceinline__ void rodrigues3(const float* r, float* R) {
    float th = sqrtf(r[0]*r[0] + r[1]*r[1] + r[2]*r[2] + 1e-8f);
    float inv = 1.0f / th;
    float nx = r[0]*inv, ny = r[1]*inv, nz = r[2]*inv;
    float c = cosf(th), s = sinf(th), mc = 1.0f - c;
    R[0] = c + mc*nx*nx;      R[1] = mc*nx*ny - s*nz;  R[2] = mc*nx*nz + s*ny;
    R[3] = mc*nx*ny + s*nz;   R[4] = c + mc*ny*ny;     R[5] = mc*ny*nz - s*nx;
    R[6] = mc*nz*nx - s*ny;   R[7] = mc*ny*nz + s*nx;  R[8] = c + mc*nz*nz;
}

__global__ __launch_bounds__(128)
void decode(const float* __restrict__ paramsWS,   // (4096, 80)
            const float* __restrict__ comps,      // (45,45)
            const float* __restrict__ meanh,      // (45,)
            const float* __restrict__ templ,      // (778,3)
            const float* __restrict__ shapedirs,  // (778,3,10)
            const float* __restrict__ posedirs,   // (778,3,135)
            const float* __restrict__ lbs,        // (778,16)
            const float* __restrict__ jbase,      // (16,3)
            const float* __restrict__ js,         // (16,3,10)
            float* __restrict__ d_out)
{
    const int token = blockIdx.x;
    const int tid = threadIdx.x;

    __shared__ float s_par[NOUT];
    __shared__ float s_pose[48];
    __shared__ float s_rot[16][9];
    __shared__ float s_J[16][3];
    __shared__ float s_res[16][12];   // rows 0..2 of 4x4 (row 3 = 0001)
    __shared__ float s_A[16][12];
    __shared__ float s_curr[5][3];
    __shared__ float s_mesh[5][3];
    __shared__ float s_pos[21][3];

    // stage 1: load params
    if (tid < NOUT) s_par[tid] = paramsWS[(size_t)token * PSTRIDE + tid];
    __syncthreads();

    // stage 2: PCA projection + global copy; J from beta (disjoint threads)
    if (tid < 3) s_pose[tid] = s_par[tid];
    if (tid < 45) {
        float acc = meanh[tid];
        for (int k = 0; k < 45; ++k)
            acc += s_par[3 + k] * comps[k * 45 + tid];
        s_pose[3 + tid] = acc;
    }
    if (tid >= 64 && tid < 64 + 48) {
        int t = tid - 64;
        int j = t / 3, c = t % 3;
        float acc = jbase[j * 3 + c];
        const float* jsrow = js + (j * 3 + c) * 10;
        for (int k = 0; k < 10; ++k)
            acc += jsrow[k] * s_par[48 + k];
        s_J[j][c] = acc;
    }
    __syncthreads();

    // stage 3: Rodrigues for 16 joints
    if (tid < 16) rodrigues3(&s_pose[3 * tid], s_rot[tid]);
    __syncthreads();

    // stage 4: root transform + per-vertex blendshape/pose corrections (5 verts)
    if (tid < 12) {
        int i = tid / 4, k = tid % 4;
        s_res[0][i * 4 + k] = (k < 3) ? s_rot[0][i * 3 + k] : s_J[0][i];
    }
    if (tid >= 32 && tid < 32 + 15) {
        const int vidx[5] = {743, 333, 443, 555, 678};
        int t = tid - 32;
        int v = t / 3, c = t % 3;
        int vi = vidx[v];
        float acc = templ[vi * 3 + c];
        const float* sd = shapedirs + vi * 30 + c * 10;
        for (int k = 0; k < 10; ++k) acc += sd[k] * s_par[48 + k];
        const float* pd = posedirs + (size_t)vi * 405 + c * 135;
        for (int k = 0; k < 135; ++k)
            acc += pd[k] * s_rot[1 + k / 9][k % 9];   // rod135 = rot16[1..15]
        s_curr[v][c] = acc;
    }
    __syncthreads();

    // stage 5: kinematic chains (5 independent chains of length 3)
    if (tid < 5) {
        int p = 0;
        int j = 3 * tid + 1;
        for (int step = 0; step < 3; ++step) {
            float rel0 = s_J[j][0] - s_J[p][0];
            float rel1 = s_J[j][1] - s_J[p][1];
            float rel2 = s_J[j][2] - s_J[p][2];
            const float* R = s_rot[j];
            const float* P = s_res[p];
            float tmp[12];
            for (int i = 0; i < 3; ++i) {
                float pi0 = P[i * 4 + 0], pi1 = P[i * 4 + 1], pi2 = P[i * 4 + 2];
                tmp[i * 4 + 0] = pi0 * R[0] + pi1 * R[3] + pi2 * R[6];
                tmp[i * 4 + 1] = pi0 * R[1] + pi1 * R[4] + pi2 * R[7];
                tmp[i * 4 + 2] = pi0 * R[2] + pi1 * R[5] + pi2 * R[8];
                tmp[i * 4 + 3] = pi0 * rel0 + pi1 * rel1 + pi2 * rel2 + P[i * 4 + 3];
            }
            for (int q = 0; q < 12; ++q) s_res[j][q] = tmp[q];
            p = j;
            ++j;
        }
    }
    __syncthreads();

    // stage 6: A = results - (R_j @ J_j) in translation column
    if (tid < 48) {
        int j = tid / 3, i = tid % 3;
        float corr = s_res[j][i * 4 + 0] * s_J[j][0]
                   + s_res[j][i * 4 + 1] * s_J[j][1]
                   + s_res[j][i * 4 + 2] * s_J[j][2];
        s_A[j][i * 4 + 0] = s_res[j][i * 4 + 0];
        s_A[j][i * 4 + 1] = s_res[j][i * 4 + 1];
        s_A[j][i * 4 + 2] = s_res[j][i * 4 + 2];
        s_A[j][i * 4 + 3] = s_res[j][i * 4 + 3] - corr;
    }
    __syncthreads();

    // stage 7: LBS at the 5 needed vertices
    if (tid < 15) {
        const int vidx[5] = {743, 333, 443, 555, 678};
        int v = tid / 3, i = tid % 3;
        int vi = vidx[v];
        float cx = s_curr[v][0], cy = s_curr[v][1], cz = s_curr[v][2];
        float acc = 0.0f;
        for (int j = 0; j < 16; ++j) {
            float w = lbs[vi * 16 + j];
            acc += w * (s_A[j][i * 4 + 0] * cx + s_A[j][i * 4 + 1] * cy +
                        s_A[j][i * 4 + 2] * cz + s_A[j][i * 4 + 3]);
        }
        s_mesh[v][i] = acc;
    }
    __syncthreads();

    // stage 8: reorder 21 keypoints (>=0 joint, <0 mesh vertex -(m+1))
    if (tid < 63) {
        const int map21[21] = {0, 13, 14, 15, -1, 1, 2, 3, -2, 4, 5, 6,
                               -3, 10, 11, 12, -4, 7, 8, 9, -5};
        int p = tid / 3, c = tid % 3;
        int m = map21[p];
        s_pos[p][c] = (m >= 0) ? s_res[m][c * 4 + 3] : s_mesh[-m - 1][c];
    }
    __syncthreads();

    // stage 9: projection + outputs
    const size_t offJ = 0;
    const size_t offT = (size_t)NTOK * 42;
    const size_t offB = (size_t)NTOK * 42 + (size_t)NTOK * 48;
    if (tid < 42) {
        int p = tid / 2, d = tid % 2;
        float val = s_pos[p][0] * s_par[58 + 0 * 3 + d]
                  + s_pos[p][1] * s_par[58 + 1 * 3 + d]
                  + s_pos[p][2] * s_par[58 + 2 * 3 + d];
        d_out[offJ + (size_t)token * 42 + tid] = s_par[69] * val + s_par[67 + d];
    }
    if (tid < 48) d_out[offT + (size_t)token * 48 + tid] = s_par[tid];
    if (tid >= 64 && tid < 74) {
        int k = tid - 64;
        d_out[offB + (size_t)token * 10 + k] = s_par[48 + k];
    }
}

// ---------------------------------------------------------------------------
extern "C" void kernel_launch(void* const* d_in, const int* in_sizes, int n_in,
                              void* d_out, int out_size, void* d_ws, size_t ws_size,
                              hipStream_t stream) {
    const float* x         = (const float*)d_in[0];
    const float* W         = (const float*)d_in[1];
    const float* b         = (const float*)d_in[2];
    const float* comps     = (const float*)d_in[3];
    const float* meanh     = (const float*)d_in[4];
    const float* templ     = (const float*)d_in[5];
    const float* shapedirs = (const float*)d_in[6];
    const float* posedirs  = (const float*)d_in[7];
    const float* J_reg     = (const float*)d_in[8];
    const float* lbs       = (const float*)d_in[9];
    float* out = (float*)d_out;

    float* wsf    = (float*)d_ws;
    float* params = wsf;                          // 4096 * 80
    float* jbase  = wsf + (size_t)NTOK * PSTRIDE; // 48
    float* js     = jbase + 48;                   // 480

    precompute_jreg<<<1, 544, 0, stream>>>(J_reg, templ, shapedirs, jbase, js);

    gemm_wmma<<<NTOK / 16, 160, 0, stream>>>(x, W, b, params);

    decode<<<NTOK, 128, 0, stream>>>(params, comps, meanh, templ, shapedirs,
                                     posedirs, lbs, jbase, js, out);
}